// FeatureScaling_42752104464652
// MI455X (gfx1250) — compile-verified
//
#include <hip/hip_runtime.h>
#include <stdint.h>

// FeatureScaling: piecewise-linear interpolation, B=8, N=2e6, F=3, K=17.
// Memory-bound streaming kernel: ~384 MB HBM traffic -> ~16.5us floor @ 23.3 TB/s.

#define FEATS 3
#define KPTS 17
#define SEGS (KPTS - 1)          // 16 segments per feature
#define TAB (FEATS * KPTS)       // 51 breakpoints total
#define SEGTAB (FEATS * SEGS)    // 48 segments total
#define CHUNK 12                 // lcm(4, FEATS): f = (elem % 3) is compile-time

typedef float v4f __attribute__((ext_vector_type(4)));
typedef float v2f __attribute__((ext_vector_type(2)));

__global__ __launch_bounds__(256)
void FeatureScaling_42752104464652_kernel(const float* __restrict__ in,
                                          const float* __restrict__ map_from,
                                          const float* __restrict__ map_to,
                                          float* __restrict__ out,
                                          long long nchunk, long long total) {
    __shared__ float s_xs[TAB];      // breakpoints (async-loaded)
    __shared__ float s_ys[TAB];      // values     (async-loaded)
    __shared__ v2f   s_ab[SEGTAB];   // per-segment {slope, intercept}

    const int t = threadIdx.x;

    // ---- Stage tables into LDS via CDNA5 async global->LDS DMA path ----
    if (t < 2 * TAB) {
        const float* gsrc = (t < TAB) ? (map_from + t) : (map_to + (t - TAB));
        float*       ldst = (t < TAB) ? (s_xs + t)     : (s_ys + (t - TAB));
        uint32_t lds_off = (uint32_t)(uintptr_t)ldst;   // low 32 bits of flat addr == LDS offset
        uint64_t gaddr   = (uint64_t)(uintptr_t)gsrc;
        asm volatile("global_load_async_to_lds_b32 %0, %1, off"
                     :: "v"(lds_off), "v"(gaddr) : "memory");
    }
    asm volatile("s_wait_asynccnt 0" ::: "memory");
    __syncthreads();

    // ---- Precompute slope/intercept per segment: y = a*x + b ----
    if (t < SEGTAB) {
        int f = t >> 4, s = t & 15;
        float x1 = s_xs[f * KPTS + s], x2 = s_xs[f * KPTS + s + 1];
        float y1 = s_ys[f * KPTS + s], y2 = s_ys[f * KPTS + s + 1];
        float a = (y2 - y1) / (x2 - x1);
        v2f ab; ab.x = a; ab.y = __builtin_fmaf(-x1, a, y1);
        s_ab[f * SEGS + t - (f << 4) + 0] = ab;   // == s_ab[f*SEGS + s]
    }
    __syncthreads();

    // ---- Interior breakpoints in SGPRs (uniform scalar loads) ----
    float xs[FEATS][SEGS];
#pragma unroll
    for (int f = 0; f < FEATS; ++f)
#pragma unroll
        for (int k = 1; k < KPTS - 1; ++k)
            xs[f][k] = map_from[f * KPTS + k];

    const long long tid     = (long long)blockIdx.x * blockDim.x + threadIdx.x;
    const long long gstride = (long long)gridDim.x * blockDim.x;

    // ---- Main streaming loop: 12 consecutive floats (4 points) per iter ----
    for (long long c = tid; c < nchunk; c += gstride) {
        const v4f* pin  = (const v4f*)(in  + c * CHUNK);
        v4f*       pout = (v4f*)(out + c * CHUNK);

        // Speculative prefetch of this thread's next chunk (safe if OOB).
        __builtin_prefetch((const void*)(in + (c + gstride) * CHUNK), 0, 0);

        v4f v0 = __builtin_nontemporal_load(pin + 0);
        v4f v1 = __builtin_nontemporal_load(pin + 1);
        v4f v2 = __builtin_nontemporal_load(pin + 2);

        float e[CHUNK] = {v0.x, v0.y, v0.z, v0.w,
                          v1.x, v1.y, v1.z, v1.w,
                          v2.x, v2.y, v2.z, v2.w};
        float r[CHUNK];
#pragma unroll
        for (int ci = 0; ci < CHUNK; ++ci) {
            const int f = ci % FEATS;          // compile-time constant
            float x = e[ci];
            int idx = 0;                       // searchsorted(side=right)-1, clamped
#pragma unroll
            for (int k = 1; k < KPTS - 1; ++k)
                idx += (x >= xs[f][k]) ? 1 : 0;
            v2f ab = s_ab[f * SEGS + idx];     // one ds_load_b64 gather
            r[ci] = __builtin_fmaf(x, ab.x, ab.y);
        }

        v4f o0 = {r[0], r[1], r[2], r[3]};
        v4f o1 = {r[4], r[5], r[6], r[7]};
        v4f o2 = {r[8], r[9], r[10], r[11]};
        __builtin_nontemporal_store(o0, pout + 0);
        __builtin_nontemporal_store(o1, pout + 1);
        __builtin_nontemporal_store(o2, pout + 2);
    }

    // ---- Tail (total not divisible by 12): LDS-indexed scalar path ----
    for (long long i = nchunk * CHUNK + tid; i < total; i += gstride) {
        float x = in[i];
        int f = (int)(i % FEATS);
        int idx = 0;
#pragma unroll
        for (int k = 1; k < KPTS - 1; ++k)
            idx += (x >= s_xs[f * KPTS + k]) ? 1 : 0;
        v2f ab = s_ab[f * SEGS + idx];
        out[i] = __builtin_fmaf(x, ab.x, ab.y);
    }
}

extern "C" void kernel_launch(void* const* d_in, const int* in_sizes, int n_in,
                              void* d_out, int out_size, void* d_ws, size_t ws_size,
                              hipStream_t stream) {
    const float* in       = (const float*)d_in[0];
    const float* map_from = (const float*)d_in[1];
    const float* map_to   = (const float*)d_in[2];
    float*       out      = (float*)d_out;

    const long long total  = (long long)in_sizes[0];   // B*N*F = 48,000,000
    const long long nchunk = total / CHUNK;            // 4,000,000

    const int block = 256;
    // ~4 chunks (192 B) per thread: deep enough for prefetch, wide enough to fill.
    long long want = (nchunk + (long long)block * 4 - 1) / ((long long)block * 4);
    if (want < 1) want = 1;
    if (want > 65535) want = 65535;
    FeatureScaling_42752104464652_kernel<<<(int)want, block, 0, stream>>>(
        in, map_from, map_to, out, nchunk, total);
}